// MASKRCNN_model_79010218377590
// MI455X (gfx1250) — compile-verified
//
#include <hip/hip_runtime.h>

// ---------------------------------------------------------------------------
// Mask R-CNN mask paste:  out[n] = Wy[n] (800x28) * mask[n] (28x28) * Wx[n]^T (28x1280)
// Bilinear separable resample expressed as WMMA f32 16x16x4 GEMM tiles (CDNA5).
// Bandwidth-bound (524 MB stores); fp32 WMMA gives exact precision for free.
// ---------------------------------------------------------------------------

typedef __attribute__((ext_vector_type(2))) float v2f;
typedef __attribute__((ext_vector_type(8))) float v8f;

#define HM 28
#define WM 28
#define IMG_H 800
#define IMG_W 1280
#define TILE_Y 16
#define WAVES 8
#define TILE_X (16 * WAVES)   // 128 columns per block

#define MASK_STRIDE (WM + 1)  // 29, avoid LDS bank conflicts
#define A_STRIDE 33           // 16 rows x K(padded 32), stride 33
#define B_STRIDE 17           // 32 k-rows x 16 cols, stride 17

__global__ __launch_bounds__(256)
void maskrcnn_paste_wmma(const float* __restrict__ masks,
                         const float* __restrict__ boxes,
                         const int* __restrict__ p_img_h,
                         const int* __restrict__ p_img_w,
                         const int* __restrict__ p_in_h,
                         const int* __restrict__ p_in_w,
                         float* __restrict__ out)
{
    __shared__ float s_mask[HM * MASK_STRIDE];          // 28 x 29
    __shared__ float s_A[TILE_Y * A_STRIDE];            // 16 x 32 (padded), Wy tile
    __shared__ float s_B[WAVES][32 * B_STRIDE];         // per-wave 32 x 16 (padded)

    const int n    = blockIdx.z;
    const int y0   = blockIdx.y * TILE_Y;
    const int x0   = blockIdx.x * TILE_X;
    const int tid  = threadIdx.x;
    const int wave = tid >> 5;        // wave32 on gfx1250
    const int lane = tid & 31;

    // --- box scale + clip (uniform per block; scalar loads) -----------------
    const float img_w_f = (float)p_img_w[0];
    const float img_h_f = (float)p_img_h[0];
    const float sx = img_w_f / (float)p_in_w[0];
    const float sy = img_h_f / (float)p_in_h[0];
    const float bx0 = fminf(fmaxf(boxes[n * 4 + 0] * sx, 0.f), img_w_f);
    const float by0 = fminf(fmaxf(boxes[n * 4 + 1] * sy, 0.f), img_h_f);
    const float bx1 = fminf(fmaxf(boxes[n * 4 + 2] * sx, 0.f), img_w_f);
    const float by1 = fminf(fmaxf(boxes[n * 4 + 3] * sy, 0.f), img_h_f);

    // --- stage mask[n] into LDS, zero the Wy tile ---------------------------
    for (int i = tid; i < HM * WM; i += 256) {
        const int r = i / WM, c = i - r * WM;
        s_mask[r * MASK_STRIDE + c] = masks[(size_t)n * (HM * WM) + i];
    }
    for (int i = tid; i < TILE_Y * A_STRIDE; i += 256) s_A[i] = 0.f;
    __syncthreads();

    // --- A = Wy tile: 16 output rows, <=2 taps each (zero padding via mask) -
    if (tid < TILE_Y) {
        const float py = (((float)(y0 + tid) + 0.5f) - by0) / (by1 - by0) * (float)HM - 0.5f;
        const float fy = floorf(py);
        const int   iy = (int)fy;
        const float w1 = py - fy;
        const float w0 = 1.f - w1;
        if (iy     >= 0 && iy     < HM) s_A[tid * A_STRIDE + iy]     = w0;
        if (iy + 1 >= 0 && iy + 1 < HM) s_A[tid * A_STRIDE + iy + 1] = w1;
    }

    // --- B = (mask * Wx^T) columns for this wave's 16-col slice -------------
    {
        const int   c     = lane & 15;
        const int   khalf = lane >> 4;                  // k base 0 or 16
        const int   x     = x0 + wave * 16 + c;
        const float px = (((float)x + 0.5f) - bx0) / (bx1 - bx0) * (float)WM - 0.5f;
        const float fx = floorf(px);
        const int   jx = (int)fx;
        const float w1 = px - fx;
        const float w0 = 1.f - w1;
        const bool  v0 = (jx     >= 0) && (jx     < WM);
        const bool  v1 = (jx + 1 >= 0) && (jx + 1 < WM);
        const int   j0 = v0 ? jx     : 0;
        const int   j1 = v1 ? jx + 1 : 0;
        const float a0 = v0 ? w0 : 0.f;
        const float a1 = v1 ? w1 : 0.f;
        float* B = &s_B[wave][0];
        #pragma unroll
        for (int kk = 0; kk < 16; ++kk) {
            const int k = khalf * 16 + kk;
            float v = 0.f;
            if (k < HM)
                v = a0 * s_mask[k * MASK_STRIDE + j0] + a1 * s_mask[k * MASK_STRIDE + j1];
            B[k * B_STRIDE + c] = v;
        }
    }
    __syncthreads();

    // --- 16x16 output tile via 8 x V_WMMA_F32_16X16X4_F32 (K = 32 padded) ---
    // A frag: lane l<16 -> A[l][k0+0..1], lane l+16 -> A[l][k0+2..3]
    // B frag: lane l<16 -> B[k0+0..1][l], lane l+16 -> B[k0+2..3][l]
    v8f acc = {};
    const int   l = lane & 15;
    const int   h = lane >> 4;
    const float* Arow = &s_A[l * A_STRIDE];
    const float* B    = &s_B[wave][0];
    #pragma unroll
    for (int s = 0; s < 8; ++s) {
        v2f a, b;
        a.x = Arow[4 * s + 2 * h + 0];
        a.y = Arow[4 * s + 2 * h + 1];
        b.x = B[(4 * s + 2 * h + 0) * B_STRIDE + l];
        b.y = B[(4 * s + 2 * h + 1) * B_STRIDE + l];
        acc = __builtin_amdgcn_wmma_f32_16x16x4_f32(
            /*neg_a=*/false, a, /*neg_b=*/false, b,
            /*c_mod=*/(short)0, acc, /*reuse_a=*/false, /*reuse_b=*/false);
    }

    // --- store D: VGPR r -> row (r + 8*h), col l ----------------------------
    const int xo = x0 + wave * 16 + l;
    float* outp = out + ((size_t)n * IMG_H + (size_t)(y0 + 8 * h)) * IMG_W + xo;
    #pragma unroll
    for (int r = 0; r < 8; ++r)
        outp[(size_t)r * IMG_W] = acc[r];
}

extern "C" void kernel_launch(void* const* d_in, const int* in_sizes, int n_in,
                              void* d_out, int out_size, void* d_ws, size_t ws_size,
                              hipStream_t stream)
{
    const float* masks  = (const float*)d_in[0];   // [N,1,28,28] f32
    const float* boxes  = (const float*)d_in[1];   // [N,4] f32
    const int*  p_img_h = (const int*)d_in[2];
    const int*  p_img_w = (const int*)d_in[3];
    const int*  p_in_h  = (const int*)d_in[4];
    const int*  p_in_w  = (const int*)d_in[5];
    float* out = (float*)d_out;

    const int N = in_sizes[0] / (HM * WM);

    dim3 grid(IMG_W / TILE_X, IMG_H / TILE_Y, N);  // 10 x 50 x 128
    dim3 block(256);                                // 8 wave32s
    maskrcnn_paste_wmma<<<grid, block, 0, stream>>>(masks, boxes,
                                                    p_img_h, p_img_w, p_in_h, p_in_w,
                                                    out);
}